// MelSpectrogram_63574105915742
// MI455X (gfx1250) — compile-verified
//
#include <hip/hip_runtime.h>
#include <hip/hip_bf16.h>
#include <math.h>
#include <stdint.h>

#define N_FFT    1024
#define HOP      256
#define N_MELS   80
#define N_BINS   513      // N_FFT/2 + 1
#define T_LEN    262144
#define N_FRAMES 1025
#define TILE_F   16
#define KPAD     516      // 4-aligned K with zero tail
#define SPAN     4864     // samples covered by one 16-frame tile (multiple of 4)

typedef __attribute__((ext_vector_type(2))) float v2f;
typedef __attribute__((ext_vector_type(8))) float v8f;

__global__ __launch_bounds__(256)
void melspec_fused_kernel(const float* __restrict__ y,
                          const float* __restrict__ melw,
                          float* __restrict__ out)
{
    __shared__ float  stage[SPAN];         // 19.0 KB raw-sample stage (async DMA)
    __shared__ float2 fb[N_FFT];           //  8.0 KB FFT scratch (re, im)
    __shared__ float  mag[TILE_F][KPAD];   // ~32.3 KB magnitude tile

    const int tid = threadIdx.x;
    const int bb  = blockIdx.y;            // batch
    const int f0  = blockIdx.x * TILE_F;   // first frame of tile
    const float* yrow = y + (size_t)bb * T_LEN;

    // ------- Stage the tile's contiguous sample span into LDS (async) -----
    // Frames f0..f0+15 need samples q in [f0*HOP-512, f0*HOP+4351].
    const int s0   = max(0, f0 * HOP - (N_FFT / 2));
    const int sEnd = min(T_LEN, f0 * HOP + (TILE_F - 1) * HOP + N_FFT - (N_FFT / 2));
    const int nS   = sEnd - s0;            // multiple of 4 (s0,sEnd are 256-aligned)
    {
        const uint32_t ldsBase = (uint32_t)(uintptr_t)(&stage[0]);
        const float*   gBase   = yrow + s0;
        for (int i = tid; i < (nS >> 2); i += 256) {
            const uint32_t ldsOff = ldsBase + (uint32_t)i * 16u;
            const float*   gp     = gBase + i * 4;
            asm volatile("global_load_async_to_lds_b128 %0, %1, off"
                         :: "v"(ldsOff), "v"(gp) : "memory");
        }
        asm volatile("s_wait_asynccnt 0x0" ::: "memory");   // my wave's DMA done
    }
    __syncthreads();                                        // all waves' DMA done

    // ---------------- Phase 1: 16 windowed FFTs -> |X| tile in LDS --------
    for (int fr = 0; fr < TILE_F; ++fr) {
        const int  f     = f0 + fr;
        const bool valid = (f < N_FRAMES);     // block-uniform
        __syncthreads();                       // fb free from previous frame
        if (valid) {
            #pragma unroll
            for (int i = 0; i < 4; ++i) {
                const int j = tid + i * 256;
                // reflect-padded sample index
                int q = f * HOP + j - (N_FFT / 2);
                q = (q < 0) ? -q : q;
                q = (q >= T_LEN) ? (2 * T_LEN - 2 - q) : q;
                const int idx = q - s0;
                const float xv = (idx >= 0 && idx < nS) ? stage[idx] : yrow[q];
                // Hann window
                const float w = 0.5f - 0.5f * __cosf(6.28318530718f * (float)j * (1.0f / 1024.0f));
                // bit-reversed scatter for DIT FFT
                const int br = (int)(__brev((unsigned)j) >> 22);
                fb[br] = make_float2(w * xv, 0.0f);
            }
        }
        __syncthreads();
        if (valid) {
            for (int s = 0; s < 10; ++s) {
                const int half = 1 << s;
                #pragma unroll
                for (int it = 0; it < 2; ++it) {
                    const int jj  = tid + it * 256;        // butterfly id 0..511
                    const int grp = jj >> s;
                    const int pos = jj & (half - 1);
                    const int i0  = (grp << (s + 1)) + pos;
                    const int i1  = i0 + half;
                    const float ang = -6.28318530718f * (float)pos / (float)(half << 1);
                    float sw, cw;
                    __sincosf(ang, &sw, &cw);
                    const float2 x0 = fb[i0];
                    const float2 x1 = fb[i1];
                    const float tr = cw * x1.x - sw * x1.y;
                    const float ti = cw * x1.y + sw * x1.x;
                    fb[i1] = make_float2(x0.x - tr, x0.y - ti);
                    fb[i0] = make_float2(x0.x + tr, x0.y + ti);
                }
                __syncthreads();
            }
        }
        // magnitudes for bins 0..512 (invalid frames -> zeros)
        for (int k = tid; k < N_BINS; k += 256) {
            const float2 X = fb[k];
            mag[fr][k] = valid ? sqrtf(X.x * X.x + X.y * X.y) : 0.0f;
        }
        if (tid < (KPAD - N_BINS)) mag[fr][N_BINS + tid] = 0.0f;  // zero K tail
    }
    __syncthreads();

    // ---------------- Phase 2: mel projection via f32 WMMA ----------------
    // C(80x16) = mel(80x513) x magT(513x16); 5 waves, one 16-row M tile each.
    const int wid  = tid >> 5;
    const int lane = tid & 31;
    if (wid < 5) {                          // wave-uniform: EXEC stays all-1s
        const int row = lane & 15;          // M row (A) / N frame (B,C)
        const int hi  = lane >> 4;          // lane half selects K pair
        const float* melRow = melw + (size_t)(wid * 16 + row) * N_BINS;
        v8f acc = {};
        for (int kb = 0; kb < 129; ++kb) {  // K = 0..515, tail zeroed
            const int k0 = kb * 4 + hi * 2;
            v2f a, b;
            a.x = (k0     < N_BINS) ? melRow[k0]     : 0.0f;
            a.y = (k0 + 1 < N_BINS) ? melRow[k0 + 1] : 0.0f;
            b.x = mag[row][k0];
            b.y = mag[row][k0 + 1];
            acc = __builtin_amdgcn_wmma_f32_16x16x4_f32(
                      false, a, false, b, (short)0, acc, false, false);
        }
        // C layout: VGPR r -> M = wid*16 + hi*8 + r, N = row; fused log10
        #pragma unroll
        for (int r = 0; r < 8; ++r) {
            const int m    = wid * 16 + hi * 8 + r;
            const int fidx = f0 + row;
            if (fidx < N_FRAMES) {
                out[((size_t)bb * N_MELS + m) * N_FRAMES + fidx] =
                    log10f(fmaxf(acc[r], 1e-5f));
            }
        }
    }
}

extern "C" void kernel_launch(void* const* d_in, const int* in_sizes, int n_in,
                              void* d_out, int out_size, void* d_ws, size_t ws_size,
                              hipStream_t stream) {
    const float* y    = (const float*)d_in[0];   // (32, 262144) f32
    const float* melw = (const float*)d_in[1];   // (80, 513)    f32
    float* out = (float*)d_out;                  // (32, 80, 1025) f32
    (void)in_sizes; (void)n_in; (void)out_size; (void)d_ws; (void)ws_size;

    dim3 grid((N_FRAMES + TILE_F - 1) / TILE_F, 32, 1);   // 65 x 32 blocks
    dim3 block(256, 1, 1);                                // 8 wave32
    hipLaunchKernelGGL(melspec_fused_kernel, grid, block, 0, stream, y, melw, out);
}